// MemexQA_FVTA_9259949490211
// MI455X (gfx1250) — compile-verified
//
#include <hip/hip_runtime.h>
#include <hip/hip_bf16.h>

typedef __attribute__((ext_vector_type(2))) float v2f;
typedef __attribute__((ext_vector_type(8))) float v8f;

// Problem constants
#define HH 2
#define BB 64
#define LTXT 384
#define LIMG 128
#define LLEN 512      // LTXT + LIMG
#define LQRY 24
#define DDIM 768
#define KDIM 384      // K == V
#define DVAL 768      // H*V == H*K == DDIM

// ---------------------------------------------------------------------------
// WMMA helpers (V_WMMA_F32_16X16X4_F32, wave32).
// A layout (16x4 f32): lanes 0-15 row=lane, VGPR0/1 = K 0/1; lanes 16-31 same
// rows, K 2/3.  B layout (4x16): lanes 0-15 col=lane, VGPR0/1 = K 0/1 rows;
// lanes 16-31 K 2/3.  C/D: VGPR i -> rows i (lanes 0-15) and i+8 (16-31).
//
// 4-way N register blocking: one wave computes a 16x64 tile (4 accumulators),
// re-using the A fragment across 4 WMMAs -> 4x less A traffic, 2x higher
// wmma:vmem density.  B stream is prefetched one K-slab ahead
// (global_prefetch_b8 path on gfx1250).
// ---------------------------------------------------------------------------
#define WMMA_F32(a, b, c) \
  __builtin_amdgcn_wmma_f32_16x16x4_f32(false, (a), false, (b), (short)0, (c), false, false)

__device__ __forceinline__ void wmma4_AB(const float* __restrict__ A, int lda,
                                         const float* __restrict__ Bm, int ldb,
                                         int kdim, int lane,
                                         v8f& c0, v8f& c1, v8f& c2, v8f& c3) {
  const int hi2 = (lane >> 4) << 1;      // 0 for lanes 0-15, 2 for 16-31
  const int r   = lane & 15;
  const float* ap = A + (size_t)r * lda; // row m0+r
  for (int k0 = 0; k0 < kdim; k0 += 4) {
    v2f a;
    a.x = ap[k0 + hi2];
    a.y = ap[k0 + hi2 + 1];
    const float* b0 = Bm + (size_t)(k0 + hi2) * ldb + r;
    const float* b1 = Bm + (size_t)(k0 + hi2 + 1) * ldb + r;
    __builtin_prefetch(b0 + 16 * (size_t)ldb, 0, 0);  // next K-slab of B
    v2f bb0{b0[0],  b1[0]};
    v2f bb1{b0[16], b1[16]};
    v2f bb2{b0[32], b1[32]};
    v2f bb3{b0[48], b1[48]};
    c0 = WMMA_F32(a, bb0, c0);
    c1 = WMMA_F32(a, bb1, c1);
    c2 = WMMA_F32(a, bb2, c2);
    c3 = WMMA_F32(a, bb3, c3);
  }
}

// Same but B accessed transposed in memory: B[k][n] = Bt[n*ldbt + k]
__device__ __forceinline__ void wmma4_ABt(const float* __restrict__ A, int lda,
                                          const float* __restrict__ Bt, int ldbt,
                                          int kdim, int lane,
                                          v8f& c0, v8f& c1, v8f& c2, v8f& c3) {
  const int hi2 = (lane >> 4) << 1;
  const int r   = lane & 15;
  const float* ap  = A  + (size_t)r * lda;
  const float* bp0 = Bt + (size_t)(r +  0) * ldbt;
  const float* bp1 = Bt + (size_t)(r + 16) * ldbt;
  const float* bp2 = Bt + (size_t)(r + 32) * ldbt;
  const float* bp3 = Bt + (size_t)(r + 48) * ldbt;
  for (int k0 = 0; k0 < kdim; k0 += 4) {
    v2f a;
    a.x = ap[k0 + hi2];
    a.y = ap[k0 + hi2 + 1];
    __builtin_prefetch(bp0 + k0 + 64, 0, 0);
    v2f bb0{bp0[k0 + hi2], bp0[k0 + hi2 + 1]};
    v2f bb1{bp1[k0 + hi2], bp1[k0 + hi2 + 1]};
    v2f bb2{bp2[k0 + hi2], bp2[k0 + hi2 + 1]};
    v2f bb3{bp3[k0 + hi2], bp3[k0 + hi2 + 1]};
    c0 = WMMA_F32(a, bb0, c0);
    c1 = WMMA_F32(a, bb1, c1);
    c2 = WMMA_F32(a, bb2, c2);
    c3 = WMMA_F32(a, bb3, c3);
  }
}

// ---------------------------------------------------------------------------
// 1. LayerNorm of text & images into fused Xn (B, L, D)
// ---------------------------------------------------------------------------
__global__ void ln_kernel(const float* __restrict__ text,
                          const float* __restrict__ images,
                          const float* __restrict__ gam,
                          const float* __restrict__ bet,
                          float* __restrict__ Xn) {
  __shared__ float red[256];
  int idx = blockIdx.x;                 // B*L rows
  int b = idx / LLEN, l = idx % LLEN;
  const float* src = (l < LTXT)
      ? text   + ((size_t)b * LTXT + l) * DDIM
      : images + ((size_t)b * LIMG + (l - LTXT)) * DDIM;
  int t = threadIdx.x;
  float x0 = src[t], x1 = src[t + 256], x2 = src[t + 512];
  red[t] = x0 + x1 + x2;
  __syncthreads();
  for (int s = 128; s > 0; s >>= 1) { if (t < s) red[t] += red[t + s]; __syncthreads(); }
  float mu = red[0] * (1.0f / 768.0f);
  __syncthreads();
  float d0 = x0 - mu, d1 = x1 - mu, d2 = x2 - mu;
  red[t] = d0 * d0 + d1 * d1 + d2 * d2;
  __syncthreads();
  for (int s = 128; s > 0; s >>= 1) { if (t < s) red[t] += red[t + s]; __syncthreads(); }
  float inv = rsqrtf(red[0] * (1.0f / 768.0f) + 1e-5f);
  float* dst = Xn + ((size_t)b * LLEN + l) * DDIM;
  dst[t]       = d0 * inv * gam[t]       + bet[t];
  dst[t + 256] = d1 * inv * gam[t + 256] + bet[t + 256];
  dst[t + 512] = d2 * inv * gam[t + 512] + bet[t + 512];
}

// ---------------------------------------------------------------------------
// 2. QKV projections: q/k/v (H,B,L,K).  One wave -> 16x64 tile.
//    waves = 3 (qkv) * 2 (h) * 2048 (mtiles) * 6 (ngroups of 64)
// ---------------------------------------------------------------------------
__global__ void proj_kernel(const float* __restrict__ Xn,
                            const float* Wsq, const float* bsq,
                            const float* Wiq, const float* biq,
                            const float* Wsk, const float* bsk,
                            const float* Wik, const float* bik,
                            const float* Wsv, const float* bsv,
                            const float* Wiv, const float* biv,
                            const int* __restrict__ tlen,
                            const int* __restrict__ ilen,
                            float* __restrict__ Qb, float* __restrict__ Kb,
                            float* __restrict__ Vb) {
  int wid  = blockIdx.x * (blockDim.x >> 5) + (threadIdx.x >> 5);
  int lane = threadIdx.x & 31;
  int ng = wid % 6;   int r1 = wid / 6;
  int mt = r1 % 2048; int r2 = r1 / 2048;
  int h  = r2 % HH;   int mat = r2 / HH;
  int m0 = mt * 16, n0 = ng * 64;
  int b = m0 / LLEN, l0 = m0 % LLEN;
  bool is_text = (l0 < LTXT);
  const float* Wt; const float* bt;
  if (mat == 0)      { Wt = is_text ? Wsq : Wiq; bt = is_text ? bsq : biq; }
  else if (mat == 1) { Wt = is_text ? Wsk : Wik; bt = is_text ? bsk : bik; }
  else               { Wt = is_text ? Wsv : Wiv; bt = is_text ? bsv : biv; }

  const float* A  = Xn + (size_t)m0 * DDIM;
  const float* Bm = Wt + (size_t)h * DDIM * KDIM + n0;
  v8f c0 = {}, c1 = {}, c2 = {}, c3 = {};
  wmma4_AB(A, DDIM, Bm, KDIM, DDIM, lane, c0, c1, c2, c3);

  int col = lane & 15, rofs = (lane >> 4) * 8;
  int   len = is_text ? tlen[b] : ilen[b];
  int lbase = is_text ? l0 : (l0 - LTXT);
  float* Out = (mat == 0 ? Qb : (mat == 1 ? Kb : Vb))
             + (size_t)h * BB * LLEN * KDIM;
  v8f cj[4] = {c0, c1, c2, c3};
#pragma unroll
  for (int j = 0; j < 4; ++j) {
    int cc = n0 + 16 * j + col;
    float bv = bt[h * KDIM + cc];
#pragma unroll
    for (int i = 0; i < 8; ++i) {
      float val = cj[j][i] + bv;
      if (lbase + rofs + i >= len) val = 0.0f;
      Out[(size_t)(m0 + rofs + i) * KDIM + cc] = val;
    }
  }
}

// ---------------------------------------------------------------------------
// 3. Attention scores w = q k^T per (h,b).  One wave -> 16x64 tile.
//    waves = 2 (h) * 64 (b) * 32 (mt) * 8 (ng)
// ---------------------------------------------------------------------------
__global__ void qk_kernel(const float* __restrict__ Qb,
                          const float* __restrict__ Kb,
                          float* __restrict__ W) {
  int wid  = blockIdx.x * (blockDim.x >> 5) + (threadIdx.x >> 5);
  int lane = threadIdx.x & 31;
  int ng = wid & 7;
  int mt = (wid >> 3) & 31;
  int b  = (wid >> 8) & 63;
  int h  = (wid >> 14) & 1;
  int m0 = mt * 16, n0 = ng * 64;
  size_t base = ((size_t)h * BB + b) * LLEN * KDIM;
  const float* A  = Qb + base + (size_t)m0 * KDIM;
  const float* Bt = Kb + base + (size_t)n0 * KDIM;
  v8f c0 = {}, c1 = {}, c2 = {}, c3 = {};
  wmma4_ABt(A, KDIM, Bt, KDIM, KDIM, lane, c0, c1, c2, c3);

  size_t wbase = ((size_t)h * BB + b) * LLEN * LLEN;
  int col = lane & 15, rofs = (lane >> 4) * 8;
  v8f cj[4] = {c0, c1, c2, c3};
#pragma unroll
  for (int j = 0; j < 4; ++j) {
#pragma unroll
    for (int i = 0; i < 8; ++i) {
      int row = m0 + rofs + i, cc = n0 + 16 * j + col;
      float val = cj[j][i];
      if (val == 0.0f || row == cc) val = -__builtin_inff();
      W[wbase + (size_t)row * LLEN + cc] = val;
    }
  }
}

// ---------------------------------------------------------------------------
// 4/8. Masked softmax over a 512-wide row (valid = finite)
// ---------------------------------------------------------------------------
__device__ __forceinline__ void softmax_row512(float* __restrict__ p,
                                               float* __restrict__ red, int t) {
  float x0 = p[t], x1 = p[t + 256];
  bool v0 = (x0 >= -3.0e38f), v1 = (x1 >= -3.0e38f);
  red[t] = fmaxf(v0 ? x0 : -3.0e38f, v1 ? x1 : -3.0e38f);
  __syncthreads();
  for (int s = 128; s > 0; s >>= 1) { if (t < s) red[t] = fmaxf(red[t], red[t + s]); __syncthreads(); }
  float wmax = red[0];
  __syncthreads();
  float e0 = v0 ? __expf(x0 - wmax) : 0.0f;
  float e1 = v1 ? __expf(x1 - wmax) : 0.0f;
  red[t] = e0 + e1;
  __syncthreads();
  for (int s = 128; s > 0; s >>= 1) { if (t < s) red[t] += red[t + s]; __syncthreads(); }
  float ssum = red[0];
  float inv = (ssum > 0.0f) ? (1.0f / ssum) : 1.0f;
  p[t] = e0 * inv;
  p[t + 256] = e1 * inv;
}

__global__ void softmax_attn_kernel(float* __restrict__ W) {
  __shared__ float red[256];
  softmax_row512(W + (size_t)blockIdx.x * LLEN, red, threadIdx.x);
}

__global__ void softmax_w2_kernel(float* __restrict__ W2) {
  __shared__ float red[256];
  int b = blockIdx.x / LQRY, q = blockIdx.x % LQRY;
  softmax_row512(W2 + ((size_t)b * 32 + q) * LLEN, red, threadIdx.x);
}

// ---------------------------------------------------------------------------
// 5. attn = p v, stored directly in album layout (B, L, H*V).
//    waves = 2 (h) * 64 (b) * 32 (mt) * 6 (ng)
// ---------------------------------------------------------------------------
__global__ void pv_kernel(const float* __restrict__ W,
                          const float* __restrict__ Vb,
                          float* __restrict__ album) {
  int wid  = blockIdx.x * (blockDim.x >> 5) + (threadIdx.x >> 5);
  int lane = threadIdx.x & 31;
  int ng = wid % 6; int r1 = wid / 6;
  int mt = r1 & 31; r1 >>= 5;
  int b  = r1 & 63; int h = r1 >> 6;
  int m0 = mt * 16, n0 = ng * 64;
  const float* A  = W  + ((size_t)h * BB + b) * LLEN * LLEN + (size_t)m0 * LLEN;
  const float* Bm = Vb + ((size_t)h * BB + b) * LLEN * KDIM + n0;
  v8f c0 = {}, c1 = {}, c2 = {}, c3 = {};
  wmma4_AB(A, LLEN, Bm, KDIM, LLEN, lane, c0, c1, c2, c3);
  int col = lane & 15, rofs = (lane >> 4) * 8;
  v8f cj[4] = {c0, c1, c2, c3};
#pragma unroll
  for (int j = 0; j < 4; ++j)
#pragma unroll
    for (int i = 0; i < 8; ++i)
      album[((size_t)b * LLEN + m0 + rofs + i) * DVAL
            + h * KDIM + n0 + 16 * j + col] = cj[j][i];
}

// ---------------------------------------------------------------------------
// 6. keys = album Wkp + bkp ; vals = album Wvp + bvp ; pad-zeroed.
//    waves = 2 (mat) * 2048 (mt) * 12 (ng)
// ---------------------------------------------------------------------------
__global__ void kvproj_kernel(const float* __restrict__ album,
                              const float* __restrict__ Wkp, const float* __restrict__ bkp,
                              const float* __restrict__ Wvp, const float* __restrict__ bvp,
                              const int* __restrict__ tlen, const int* __restrict__ ilen,
                              float* __restrict__ keys, float* __restrict__ vals) {
  int wid  = blockIdx.x * (blockDim.x >> 5) + (threadIdx.x >> 5);
  int lane = threadIdx.x & 31;
  int ng = wid % 12; int r1 = wid / 12;
  int mt = r1 % 2048; int mat = r1 / 2048;
  int m0 = mt * 16, n0 = ng * 64;
  const float* A  = album + (size_t)m0 * DVAL;
  const float* Bm = (mat ? Wvp : Wkp) + n0;
  v8f c0 = {}, c1 = {}, c2 = {}, c3 = {};
  wmma4_AB(A, DVAL, Bm, DVAL, DVAL, lane, c0, c1, c2, c3);
  int col = lane & 15, rofs = (lane >> 4) * 8;
  const float* bm = mat ? bvp : bkp;
  float* Out = mat ? vals : keys;
  v8f cj[4] = {c0, c1, c2, c3};
#pragma unroll
  for (int j = 0; j < 4; ++j) {
    int cc = n0 + 16 * j + col;
    float bv = bm[cc];
#pragma unroll
    for (int i = 0; i < 8; ++i) {
      int m = m0 + rofs + i;
      int b = m / LLEN, l = m % LLEN;
      bool pad = (l < LTXT) ? (l >= tlen[b]) : ((l - LTXT) >= ilen[b]);
      float val = cj[j][i] + bv;
      if (pad) val = 0.0f;
      Out[(size_t)m * DVAL + cc] = val;
    }
  }
}

// ---------------------------------------------------------------------------
// 7. w2 = qf keys^T (M padded to 32; invalid rows contribute zeros)
//    waves = 64 (b) * 2 (mt) * 32 (nt)   -- small, single-tile waves
// ---------------------------------------------------------------------------
__global__ void w2_kernel(const float* __restrict__ query,
                          const float* __restrict__ keys,
                          const int* __restrict__ qlen,
                          float* __restrict__ W2) {
  int wid  = blockIdx.x * (blockDim.x >> 5) + (threadIdx.x >> 5);
  int lane = threadIdx.x & 31;
  int nt = wid & 31;
  int mt = (wid >> 5) & 1;
  int b  = wid >> 6;
  int m0 = mt * 16, n0 = nt * 16;
  int r = lane & 15, hi2 = (lane >> 4) << 1;
  int rowq = m0 + r;
  bool ok = (rowq < LQRY) && (rowq < qlen[b]);
  const float* ap = query + ((size_t)b * LQRY + (rowq < LQRY ? rowq : 0)) * DVAL;
  const float* bp = keys  + ((size_t)b * LLEN + n0 + r) * DVAL;
  v8f acc = {};
  for (int k0 = 0; k0 < DVAL; k0 += 4) {
    v2f a, bb;
    a.x = ok ? ap[k0 + hi2]     : 0.0f;
    a.y = ok ? ap[k0 + hi2 + 1] : 0.0f;
    bb.x = bp[k0 + hi2];
    bb.y = bp[k0 + hi2 + 1];
    acc = WMMA_F32(a, bb, acc);
  }
  int col = lane & 15, rofs = (lane >> 4) * 8;
#pragma unroll
  for (int i = 0; i < 8; ++i) {
    float val = acc[i];
    if (val == 0.0f) val = -__builtin_inff();
    W2[((size_t)b * 32 + m0 + rofs + i) * LLEN + n0 + col] = val;
  }
}

// ---------------------------------------------------------------------------
// 9. c[b,k] = mean over q<24 of p2[b,q,k]
// ---------------------------------------------------------------------------
__global__ void mean_kernel(const float* __restrict__ W2, float* __restrict__ cbuf) {
  int idx = blockIdx.x * blockDim.x + threadIdx.x;   // B*512
  int b = idx / LLEN, kk = idx % LLEN;
  float s = 0.0f;
  for (int q = 0; q < LQRY; ++q)
    s += W2[((size_t)b * 32 + q) * LLEN + kk];
  cbuf[idx] = s * (1.0f / (float)LQRY);
}

// ---------------------------------------------------------------------------
// 10. out[b,v] = sum_k c[b,k] * vals[b,k,v]
// ---------------------------------------------------------------------------
__global__ void out_kernel(const float* __restrict__ cbuf,
                           const float* __restrict__ vals,
                           float* __restrict__ out) {
  int idx = blockIdx.x * blockDim.x + threadIdx.x;   // B*768
  int b = idx / DVAL, vcol = idx % DVAL;
  const float* vp = vals + (size_t)b * LLEN * DVAL + vcol;
  const float* cp = cbuf + (size_t)b * LLEN;
  float s = 0.0f;
  for (int kk = 0; kk < LLEN; ++kk)
    s += cp[kk] * vp[(size_t)kk * DVAL];
  out[idx] = s;
}

// ---------------------------------------------------------------------------
extern "C" void kernel_launch(void* const* d_in, const int* in_sizes, int n_in,
                              void* d_out, int out_size, void* d_ws, size_t ws_size,
                              hipStream_t stream) {
  const float* text   = (const float*)d_in[0];
  const float* images = (const float*)d_in[1];
  const float* query  = (const float*)d_in[2];
  const float* gam    = (const float*)d_in[3];
  const float* bet    = (const float*)d_in[4];
  const float* Wsq = (const float*)d_in[5];  const float* bsq = (const float*)d_in[6];
  const float* Wiq = (const float*)d_in[7];  const float* biq = (const float*)d_in[8];
  const float* Wsk = (const float*)d_in[9];  const float* bsk = (const float*)d_in[10];
  const float* Wik = (const float*)d_in[11]; const float* bik = (const float*)d_in[12];
  const float* Wsv = (const float*)d_in[13]; const float* bsv = (const float*)d_in[14];
  const float* Wiv = (const float*)d_in[15]; const float* biv = (const float*)d_in[16];
  const float* Wkp = (const float*)d_in[17]; const float* bkp = (const float*)d_in[18];
  const float* Wvp = (const float*)d_in[19]; const float* bvp = (const float*)d_in[20];
  const int* tlen = (const int*)d_in[21];
  const int* ilen = (const int*)d_in[22];
  const int* qlen = (const int*)d_in[23];
  float* out = (float*)d_out;

  float* ws = (float*)d_ws;
  const size_t SZ = (size_t)BB * LLEN * DDIM;          // 25,165,824 floats
  float* Xn    = ws;                                    // (B,L,D)
  float* Qb    = ws + SZ;                               // (H,B,L,K) == SZ
  float* Kb    = ws + 2 * SZ;
  float* Vb    = ws + 3 * SZ;
  float* W     = ws + 4 * SZ;                           // (H,B,L,L) = 33,554,432
  float* album = Xn;                                    // Xn dead after proj
  float* keys  = Qb;                                    // q dead after qk
  float* vals  = Kb;                                    // k dead after qk
  float* W2    = W + (size_t)HH * BB * LLEN * LLEN;     // (B,32,512)
  float* cbuf  = W2 + (size_t)BB * 32 * LLEN;           // (B,512)

  ln_kernel<<<BB * LLEN, 256, 0, stream>>>(text, images, gam, bet, Xn);
  proj_kernel<<<9216, 256, 0, stream>>>(Xn, Wsq, bsq, Wiq, biq, Wsk, bsk,
                                        Wik, bik, Wsv, bsv, Wiv, biv,
                                        tlen, ilen, Qb, Kb, Vb);
  qk_kernel<<<4096, 256, 0, stream>>>(Qb, Kb, W);
  softmax_attn_kernel<<<HH * BB * LLEN, 256, 0, stream>>>(W);
  pv_kernel<<<3072, 256, 0, stream>>>(W, Vb, album);
  kvproj_kernel<<<6144, 256, 0, stream>>>(album, Wkp, bkp, Wvp, bvp,
                                          tlen, ilen, keys, vals);
  w2_kernel<<<512, 256, 0, stream>>>(query, keys, qlen, W2);
  softmax_w2_kernel<<<BB * LQRY, 256, 0, stream>>>(W2);
  mean_kernel<<<(BB * LLEN) / 256, 256, 0, stream>>>(W2, cbuf);
  out_kernel<<<(BB * DVAL) / 256, 256, 0, stream>>>(cbuf, vals, out);
}